// EdgeTransformerLayer_24318104830511
// MI455X (gfx1250) — compile-verified
//
#include <hip/hip_runtime.h>
#include <hip/hip_bf16.h>
#include <math.h>

// Problem dims (B=1)
#define NN     96            // sequence length (both edge dims)
#define DD     128           // IN_DIM == EMBED
#define HEADS  4
#define HD     32            // head dim == wave32 width
#define MROWS  (NN * NN)     // 9216 flattened (x,y) rows
#define LN_EPS 1e-5f

typedef __attribute__((ext_vector_type(2))) float v2f;
typedef __attribute__((ext_vector_type(8))) float v8f;

// ---------------- wave32 reductions (lower to ds_swizzle / permlane) -------
__device__ __forceinline__ float wave_sum(float v) {
#pragma unroll
  for (int m = 16; m >= 1; m >>= 1) v += __shfl_xor(v, m, 32);
  return v;
}
__device__ __forceinline__ float wave_max(float v) {
#pragma unroll
  for (int m = 16; m >= 1; m >>= 1) v = fmaxf(v, __shfl_xor(v, m, 32));
  return v;
}

// ---------------------------------------------------------------------------
// Kernel 1: p = x @ W + b   (M=9216, K=128, N=128) via V_WMMA_F32_16X16X4_F32
// Block = 256 threads (8 waves). Wave w computes the 16x16 tile at
// (rows blockIdx.x*16, cols w*16).
// A 16x4 f32 frag: lanes 0-15 hold row (lane&15), K = k0+0 / k0+1;
//                  lanes 16-31 hold same rows,  K = k0+2 / k0+3.
// B 4x16 f32 frag: VGPR v holds row k0+v (lanes 0-15) / k0+v+2 (lanes 16-31),
//                  col = lane&15.
// C/D:             VGPR v -> row v + 8*(lane>>4), col = lane&15.
// ---------------------------------------------------------------------------
__global__ void proj_wmma(const float* __restrict__ x, const float* __restrict__ W,
                          const float* __restrict__ bias, float* __restrict__ p) {
  const int lane = threadIdx.x & 31;
  const int wave = threadIdx.x >> 5;      // 0..7 -> column tile
  const int m0   = blockIdx.x * 16;
  const int n0   = wave * 16;
  const int half = lane >> 4;             // 0 or 1
  const int l    = lane & 15;

  const float* __restrict__ arow = x + (size_t)(m0 + l) * DD;
  v8f acc = {};
#pragma unroll 4
  for (int k0 = 0; k0 < DD; k0 += 4) {
    const int ka = k0 + 2 * half;
    v2f a = *(const v2f*)(arow + ka);                 // b64 load, K=ka, ka+1
    v2f b;
    b.x = W[(size_t)(ka + 0) * DD + n0 + l];
    b.y = W[(size_t)(ka + 1) * DD + n0 + l];
    acc = __builtin_amdgcn_wmma_f32_16x16x4_f32(false, a, false, b,
                                                (short)0, acc, false, false);
  }
  const float bv = bias[n0 + l];
#pragma unroll
  for (int v = 0; v < 8; ++v) {
    const int row = m0 + v + 8 * half;
    p[(size_t)row * DD + n0 + l] = acc[v] + bv;
  }
}

// ---------------------------------------------------------------------------
// Kernel 2: pn = LayerNorm(p) over each 32-wide head. One wave per (row,head).
// Block = 128 threads = 4 waves = 4 heads of one row.
// ---------------------------------------------------------------------------
__global__ void head_ln(const float* __restrict__ p, const float* __restrict__ g,
                        const float* __restrict__ b, float* __restrict__ pn) {
  const int m = blockIdx.x;               // 0..9215
  const int t = threadIdx.x;              // h = t>>5, d = t&31
  const int d = t & 31;
  const float v  = p[(size_t)m * DD + t];
  const float s  = wave_sum(v);
  const float s2 = wave_sum(v * v);
  const float mu  = s * (1.0f / HD);
  const float var = s2 * (1.0f / HD) - mu * mu;
  const float r   = rsqrtf(var + LN_EPS);
  pn[(size_t)m * DD + t] = (v - mu) * r * g[d] + b[d];
}

// ---------------------------------------------------------------------------
// Kernel 3: fused edge attention. One wave per (x,y,h); lane == d (HD==32).
//   pass 1: s[a] = <pn[x,a,h,:], pn[a,y,h,:]> / sqrt(32)   (wave dot product)
//   softmax over a held entirely in registers (3 scores per lane)
//   pass 2: agg[d] = sum_a att[a] * LN_d(p[x,a,h,:] * p[a,y,h,:])
// ---------------------------------------------------------------------------
__global__ void edge_attention(const float* __restrict__ p, const float* __restrict__ pn,
                               const float* __restrict__ g, const float* __restrict__ b,
                               float* __restrict__ agg) {
  const int xy = blockIdx.x;              // 0..9215
  const int xr = xy / NN;
  const int yr = xy % NN;
  const int h  = threadIdx.x >> 5;
  const int d  = threadIdx.x & 31;

  const float* __restrict__ qbase = pn + (size_t)(xr * NN) * DD + h * HD; // +a*DD+d
  const float* __restrict__ kbase = pn + (size_t)yr * DD + h * HD;        // +a*NN*DD+d
  const float inv_sqrt_hd = 0.17677669529663687f;   // 1/sqrt(32)

  // ---- pass 1: scores, 3 per lane (a = i*32 + lane) ----
  float sreg[3] = {0.f, 0.f, 0.f};
  for (int a = 0; a < NN; ++a) {
    const float qa = qbase[(size_t)a * DD + d];
    const float ka = kbase[(size_t)a * NN * DD + d];
    const float s  = wave_sum(qa * ka) * inv_sqrt_hd;
    if ((a & 31) == d) sreg[a >> 5] = s;
  }
  // ---- softmax over a (96 values across the wave) ----
  float mx = fmaxf(sreg[0], fmaxf(sreg[1], sreg[2]));
  mx = wave_max(mx);
  float e0 = __expf(sreg[0] - mx);
  float e1 = __expf(sreg[1] - mx);
  float e2 = __expf(sreg[2] - mx);
  const float inv_den = 1.0f / wave_sum(e0 + e1 + e2);
  float eArr[3] = {e0, e1, e2};

  // ---- pass 2: weighted, normalized product aggregation ----
  const float gd = g[d], bd = b[d];
  const float* __restrict__ pqb = p + (size_t)(xr * NN) * DD + h * HD;
  const float* __restrict__ pkb = p + (size_t)yr * DD + h * HD;
  float acc = 0.0f;
  for (int a = 0; a < NN; ++a) {
    const float att = __shfl(eArr[a >> 5], a & 31, 32) * inv_den;
    const float pq = pqb[(size_t)a * DD + d];
    const float pk = pkb[(size_t)a * NN * DD + d];
    const float v  = pq * pk;
    const float s  = wave_sum(v);
    const float s2 = wave_sum(v * v);
    const float mu  = s * (1.0f / HD);
    const float var = s2 * (1.0f / HD) - mu * mu;
    const float r   = rsqrtf(var + LN_EPS);
    acc = fmaf(att, (v - mu) * r * gd + bd, acc);
  }
  agg[(size_t)xy * DD + h * HD + d] = acc;
}

// ---------------------------------------------------------------------------
// Kernel 4: out_raw = [x | agg] @ Wf + bf   (M=9216, K=256, N=128), WMMA f32.
// ---------------------------------------------------------------------------
__global__ void final_wmma(const float* __restrict__ x, const float* __restrict__ agg,
                           const float* __restrict__ Wf, const float* __restrict__ bf,
                           float* __restrict__ outraw) {
  const int lane = threadIdx.x & 31;
  const int wave = threadIdx.x >> 5;
  const int m0   = blockIdx.x * 16;
  const int n0   = wave * 16;
  const int half = lane >> 4;
  const int l    = lane & 15;

  const float* __restrict__ arow0 = x   + (size_t)(m0 + l) * DD;   // k < 128
  const float* __restrict__ arow1 = agg + (size_t)(m0 + l) * DD;   // k >= 128
  v8f acc = {};
#pragma unroll 4
  for (int k0 = 0; k0 < 2 * DD; k0 += 4) {
    const int kk = k0 + 2 * half;
    const float* src = (kk < DD) ? (arow0 + kk) : (arow1 + (kk - DD));
    v2f a = *(const v2f*)src;
    v2f b;
    b.x = Wf[(size_t)(kk + 0) * DD + n0 + l];
    b.y = Wf[(size_t)(kk + 1) * DD + n0 + l];
    acc = __builtin_amdgcn_wmma_f32_16x16x4_f32(false, a, false, b,
                                                (short)0, acc, false, false);
  }
  const float bv = bf[n0 + l];
#pragma unroll
  for (int v = 0; v < 8; ++v) {
    const int row = m0 + v + 8 * half;
    outraw[(size_t)row * DD + n0 + l] = acc[v] + bv;
  }
}

// ---------------------------------------------------------------------------
// Kernel 5: final LayerNorm over the full 128-wide embedding. Wave per row,
// 4 elements per lane.
// ---------------------------------------------------------------------------
__global__ void final_ln(const float* __restrict__ outraw, const float* __restrict__ g,
                         const float* __restrict__ b, float* __restrict__ out) {
  const int m    = blockIdx.x * 4 + (threadIdx.x >> 5);
  const int lane = threadIdx.x & 31;
  float v[4];
  float s = 0.f, s2 = 0.f;
#pragma unroll
  for (int j = 0; j < 4; ++j) {
    v[j] = outraw[(size_t)m * DD + lane + 32 * j];
    s += v[j];
    s2 = fmaf(v[j], v[j], s2);
  }
  s  = wave_sum(s);
  s2 = wave_sum(s2);
  const float mu  = s * (1.0f / DD);
  const float var = s2 * (1.0f / DD) - mu * mu;
  const float r   = rsqrtf(var + LN_EPS);
#pragma unroll
  for (int j = 0; j < 4; ++j) {
    const int e = lane + 32 * j;
    out[(size_t)m * DD + e] = (v[j] - mu) * r * g[e] + b[e];
  }
}

// ---------------------------------------------------------------------------
extern "C" void kernel_launch(void* const* d_in, const int* in_sizes, int n_in,
                              void* d_out, int out_size, void* d_ws, size_t ws_size,
                              hipStream_t stream) {
  const float* x    = (const float*)d_in[0];   // (1,96,96,128)
  const float* W    = (const float*)d_in[1];   // (128,128)
  const float* bias = (const float*)d_in[2];   // (128)
  const float* ln_g = (const float*)d_in[3];   // (32)
  const float* ln_b = (const float*)d_in[4];   // (32)
  const float* Wf   = (const float*)d_in[5];   // (256,128)
  const float* bf   = (const float*)d_in[6];   // (128)
  const float* g2   = (const float*)d_in[7];   // (128)
  const float* b2   = (const float*)d_in[8];   // (128)
  float* out = (float*)d_out;                  // (1,96,96,128)

  // Workspace layout: [ p | pn | agg ], out_raw aliases p (p dead by then).
  const size_t slab = (size_t)MROWS * DD;      // 9216*128 floats = 4.72 MB
  float* p      = (float*)d_ws;
  float* pn     = p + slab;
  float* agg    = pn + slab;
  float* outraw = p;                           // reuse

  proj_wmma     <<<MROWS / 16, 256, 0, stream>>>(x, W, bias, p);
  head_ln       <<<MROWS, 128, 0, stream>>>(p, ln_g, ln_b, pn);
  edge_attention<<<MROWS, 128, 0, stream>>>(p, pn, ln_g, ln_b, agg);
  final_wmma    <<<MROWS / 16, 256, 0, stream>>>(x, agg, Wf, bf, outraw);
  final_ln      <<<MROWS / 4, 128, 0, stream>>>(outraw, g2, b2, out);
}